// InternalGraphConvolutionLayer_36112085025451
// MI455X (gfx1250) — compile-verified
//
#include <hip/hip_runtime.h>
#include <stdint.h>

typedef float v2f __attribute__((ext_vector_type(2)));
typedef float v8f __attribute__((ext_vector_type(8)));

#define D 128
#define KB 32          // K-panel staged in LDS per iteration
#define LSTR 34        // padded k-stride (dwords): 34*n mod 64 distinct for n=0..15

// Fused dual GEMM: xw = x@W, xm = x@M using V_WMMA_F32_16X16X4_F32.
// Block = 256 threads = 8 waves; each wave owns a 16-row strip, all 128 cols.
__global__ __launch_bounds__(256) void gcn_dual_gemm_wmma(
    const float* __restrict__ x, const float* __restrict__ W,
    const float* __restrict__ M, float* __restrict__ xw,
    float* __restrict__ xm, int nNodes)
{
    __shared__ float sW[D * LSTR];   // transposed [n][k] panel of W
    __shared__ float sM[D * LSTR];   // transposed [n][k] panel of M

    const int tid  = threadIdx.x;
    const int wave = tid >> 5;
    const int lane = tid & 31;
    const int mrow = lane & 15;           // A row / B col within tile
    const int kk   = (lane >> 4) << 1;    // K-pair select: 0 or 2
    const int r0   = blockIdx.x * 128 + wave * 16;

    v8f accW[8], accM[8];
#pragma unroll
    for (int ct = 0; ct < 8; ++ct) {
        accW[ct] = (v8f)0.0f;
        accM[ct] = (v8f)0.0f;
    }

    int rowA = r0 + mrow;
    if (rowA > nNodes - 1) rowA = nNodes - 1;      // clamp: WMMA needs EXEC all-1s
    const float* __restrict__ xrow = x + (long)rowA * D;

    for (int kb = 0; kb < D; kb += KB) {
        __syncthreads();
        // Cooperative stage of W/M k-panels into LDS, transposed to [n][k].
#pragma unroll
        for (int i = 0; i < (KB * D) / 256; ++i) {
            int idx = tid + i * 256;
            int k = idx >> 7;        // 0..31
            int n = idx & 127;       // 0..127
            sW[n * LSTR + k] = W[(kb + k) * D + n];
            sM[n * LSTR + k] = M[(kb + k) * D + n];
        }
        __syncthreads();

#pragma unroll
        for (int k8 = 0; k8 < KB; k8 += 4) {
            // A operand: 16x4 f32 tile, per-lane float2 (K-pair)
            v2f a = *(const v2f*)(xrow + kb + k8 + kk);
#pragma unroll
            for (int ct = 0; ct < 8; ++ct) {
                const int n = ct * 16 + mrow;
                v2f bw = *(const v2f*)(&sW[n * LSTR + k8 + kk]);  // ds_load_b64
                v2f bm = *(const v2f*)(&sM[n * LSTR + k8 + kk]);
                accW[ct] = __builtin_amdgcn_wmma_f32_16x16x4_f32(
                    false, a, false, bw, (short)0, accW[ct], false, false);
                accM[ct] = __builtin_amdgcn_wmma_f32_16x16x4_f32(
                    false, a, false, bm, (short)0, accM[ct], false, false);
            }
        }
    }

    // C/D layout: VGPR v -> row r0 + v + 8*(lane>>4), col = ct*16 + (lane&15)
    const int rowHi = (lane >> 4) << 3;
#pragma unroll
    for (int ct = 0; ct < 8; ++ct) {
        const int col = ct * 16 + mrow;
#pragma unroll
        for (int v = 0; v < 8; ++v) {
            const int row = r0 + rowHi + v;
            if (row < nNodes) {
                xw[(long)row * D + col] = accW[ct][v];
                xm[(long)row * D + col] = accM[ct][v];
            }
        }
    }
}

// Edge scatter: out[dst] += xm[src]. 32 lanes per edge, 4 floats per lane.
// All RMW traffic stays resident in the 192MB L2 (out is 25.6MB).
__global__ __launch_bounds__(256) void gcn_edge_scatter(
    const float* __restrict__ xm, const long long* __restrict__ edges,
    float* __restrict__ out, int nEdges)
{
    long long gid = (long long)blockIdx.x * 256 + threadIdx.x;
    int e = (int)(gid >> 5);
    if (e >= nEdges) return;
    int lane = (int)(gid & 31);

    long long s = edges[e];                    // edges[0][e]
    long long d = edges[(long long)nEdges + e]; // edges[1][e]

    const float* __restrict__ sp = xm + s * D + lane * 4;
    float* __restrict__ dp = out + d * D + lane * 4;
    float4 v = *(const float4*)sp;
    atomicAdd(dp + 0, v.x);
    atomicAdd(dp + 1, v.y);
    atomicAdd(dp + 2, v.z);
    atomicAdd(dp + 3, v.w);
}

__global__ __launch_bounds__(256) void gcn_relu_inplace(float* __restrict__ out, int n4)
{
    int i = blockIdx.x * 256 + threadIdx.x;
    if (i < n4) {
        float4* p = (float4*)out + i;
        float4 v = *p;
        v.x = fmaxf(v.x, 0.0f);
        v.y = fmaxf(v.y, 0.0f);
        v.z = fmaxf(v.z, 0.0f);
        v.w = fmaxf(v.w, 0.0f);
        *p = v;
    }
}

extern "C" void kernel_launch(void* const* d_in, const int* in_sizes, int n_in,
                              void* d_out, int out_size, void* d_ws, size_t ws_size,
                              hipStream_t stream) {
    const float* x      = (const float*)d_in[0];
    const float* W      = (const float*)d_in[1];
    const float* M      = (const float*)d_in[2];
    const long long* ed = (const long long*)d_in[3];
    float* out          = (float*)d_out;
    float* xm           = (float*)d_ws;   // 50000*128*4 = 25.6MB scratch

    const int nNodes = in_sizes[0] / D;
    const int nEdges = in_sizes[3] / 2;

    // 1) xw -> d_out (pre-activation), xm -> workspace (one fused WMMA pass)
    int gemmBlocks = (nNodes + 127) / 128;
    gcn_dual_gemm_wmma<<<gemmBlocks, 256, 0, stream>>>(x, W, M, out, xm, nNodes);

    // 2) out[dst] += xm[src] over all edges (L2-resident f32 atomics)
    long long sThreads = (long long)nEdges * 32;
    int sBlocks = (int)((sThreads + 255) / 256);
    gcn_edge_scatter<<<sBlocks, 256, 0, stream>>>(xm, ed, out, nEdges);

    // 3) ReLU in place
    int n4 = (nNodes * D) / 4;
    gcn_relu_inplace<<<(n4 + 255) / 256, 256, 0, stream>>>(out, n4);
}